// ConexaoRegional_15753940042199
// MI455X (gfx1250) — compile-verified
//
#include <hip/hip_runtime.h>

// ConexaoRegional: out[b,n,ds] = sum_k x_region[b,n,k] * pesos[n,ds,k]
// Per-region GEMM [32 x 16] x [16 x 256] via V_WMMA_F32_16X16X4_F32.
// Memory-bound problem (~200 MiB traffic, ~8.6us floor @ 23.3 TB/s), so the
// layout is chosen for coalesced stores on the 128 MiB output stream:
// WMMA N-axis = ds (the contiguous output axis, lane-striped in C/D layout).

typedef __attribute__((ext_vector_type(2))) float v2f;
typedef __attribute__((ext_vector_type(8))) float v8f;

#define GW     64       // regions per image row (W / R)
#define WROW   256      // W
#define HW     65536    // H * W (per-batch image stride)
#define PES_N  4096     // floats per region in pesos (D*S*R*R)
#define NDS    256      // D * S (contiguous output axis)
#define OUT_B  1048576  // N * D * S (per-batch output stride)

__global__ __launch_bounds__(256)
void conexao_regional_wmma(const float* __restrict__ x,
                           const float* __restrict__ pesos,
                           float* __restrict__ out) {
  const int n    = blockIdx.x;   // region id 0..4095
  const int i    = n >> 6;       // region row
  const int j    = n & 63;       // region col
  const int tid  = threadIdx.x;
  const int wave = tid >> 5;     // 0..7 (wave32)
  const int lane = tid & 31;
  const int half = lane >> 4;    // K-pair select: lanes 0-15 -> K{0,1}, 16-31 -> K{2,3}
  const int lid  = lane & 15;

  // Region top-left in x, pre-offset by the per-half K columns (w = half*2).
  const float* xbase = x + (size_t)(i * 4) * WROW + (size_t)(j * 4) + half * 2;
  // pesos[n] is a contiguous [256][16] block, K contiguous.
  const float* pbase = pesos + (size_t)n * PES_N + half * 2;

  // ---- A fragments: x region, M = batch (2 tiles of 16), K = 16 in 4 chunks of 4.
  // 16x4 f32 A layout: lane<16 holds M=lid,K={kk*4+0,kk*4+1}; lane>=16 K={+2,+3}.
  v2f a[2][4];
#pragma unroll
  for (int m = 0; m < 2; ++m) {
    const int b = m * 16 + lid;  // batch index
#pragma unroll
    for (int kk = 0; kk < 4; ++kk) {  // kk == region row h; half*2 == region col w
      a[m][kk] = *(const v2f*)(xbase + (size_t)b * HW + (size_t)kk * WROW);
    }
  }

  // ---- Each wave owns 2 ds-tiles; computes both batch tiles for each.
#pragma unroll
  for (int tt = 0; tt < 2; ++tt) {
    const int t  = wave * 2 + tt;   // ds tile 0..15
    const int ds = t * 16 + lid;    // this lane's output column (N axis)

    // B fragments: pesos^T, B[k][ds] = pesos[n][ds][k].
    // 4x16 f32 B layout mirrors A: lane<16 rows K{kk*4+0,+1}, lane>=16 K{+2,+3}.
    v2f bfrag[4];
#pragma unroll
    for (int kk = 0; kk < 4; ++kk) {
      bfrag[kk] = *(const v2f*)(pbase + (size_t)ds * 16 + kk * 4);
    }

    v8f c0 = {};  // batch tile 0 (b = 0..15)
    v8f c1 = {};  // batch tile 1 (b = 16..31)
#pragma unroll
    for (int kk = 0; kk < 4; ++kk) {
      c0 = __builtin_amdgcn_wmma_f32_16x16x4_f32(
          /*neg_a=*/false, a[0][kk], /*neg_b=*/false, bfrag[kk],
          /*c_mod=*/(short)0, c0, /*reuse_a=*/false, /*reuse_b=*/false);
      c1 = __builtin_amdgcn_wmma_f32_16x16x4_f32(
          /*neg_a=*/false, a[1][kk], /*neg_b=*/false, bfrag[kk],
          /*c_mod=*/(short)0, c1, /*reuse_a=*/false, /*reuse_b=*/false);
    }

    // ---- Store. C/D layout: VGPR v -> M = v (lanes 0-15) or v+8 (lanes 16-31),
    // N = lane%16 = ds. Consecutive lanes hit consecutive ds -> 64B segments.
    float* obase = out + (size_t)n * NDS + ds;
#pragma unroll
    for (int v = 0; v < 8; ++v) {
      const int b0 = v + half * 8;                 // batch within tile
      obase[(size_t)b0 * OUT_B]        = c0[v];    // batch tile 0
      obase[(size_t)(b0 + 16) * OUT_B] = c1[v];    // batch tile 1
    }
  }
}

extern "C" void kernel_launch(void* const* d_in, const int* in_sizes, int n_in,
                              void* d_out, int out_size, void* d_ws, size_t ws_size,
                              hipStream_t stream) {
  const float* x     = (const float*)d_in[0];   // [32,1,256,256] f32
  const float* pesos = (const float*)d_in[1];   // [4096,16,16,4,4] f32
  float*       out   = (float*)d_out;           // [32,4096,16,16] f32
  (void)in_sizes; (void)n_in; (void)out_size; (void)d_ws; (void)ws_size;

  dim3 grid(4096);   // one workgroup per region
  dim3 block(256);   // 8 waves (wave32)
  hipLaunchKernelGGL(conexao_regional_wmma, grid, block, 0, stream, x, pesos, out);
}